// Up_74698071212032
// MI455X (gfx1250) — compile-verified
//
#include <hip/hip_runtime.h>

#define NV   40962
#define NF   81920
#define NVP  10242
#define BB   2
#define CH   64
#define OC   64
#define EPSV 1e-5f
#define MTILES ((NV + 15) / 16)
#define WPB  8   // waves per block (256 threads)

typedef __attribute__((ext_vector_type(2))) float v2f;
typedef __attribute__((ext_vector_type(8))) float v8f;

// ---------------------------------------------------------------- utilities
__global__ void fill_ones_kernel(float* __restrict__ p, int n) {
  int i = blockIdx.x * blockDim.x + threadIdx.x;
  if (i < n) p[i] = 1.0f;
}

__global__ void scatter_x1_kernel(const float* __restrict__ x1,
                                  const int* __restrict__ v2p,
                                  float* __restrict__ xin) {
  int i = blockIdx.x * blockDim.x + threadIdx.x;
  const int total = BB * CH * NVP;
  if (i >= total) return;
  int v  = i % NVP;
  int bc = i / NVP;
  xin[(long)bc * NV + v2p[v]] = x1[i];
}

// gf = spmm(x, G) reshaped (B,C,3,NF); fused dot with EW and NS per face
__global__ void face_grad_kernel(const float* __restrict__ x,
                                 const int* __restrict__ Gc,
                                 const float* __restrict__ Gv,
                                 const float* __restrict__ EW,
                                 const float* __restrict__ NS,
                                 float* __restrict__ gde,
                                 float* __restrict__ gdn) {
  int i = blockIdx.x * blockDim.x + threadIdx.x;
  const int total = BB * CH * NF;
  if (i >= total) return;
  int f  = i % NF;
  int bc = i / NF;
  const float* xb = x + (long)bc * NV;
  float e = 0.f, nn = 0.f;
#pragma unroll
  for (int j = 0; j < 3; ++j) {
    long row = (long)j * NF + f;
    float acc = 0.f;
#pragma unroll
    for (int k = 0; k < 3; ++k)
      acc += xb[Gc[row * 3 + k]] * Gv[row * 3 + k];
    e  += acc * EW[f * 3 + j];
    nn += acc * NS[f * 3 + j];
  }
  gde[i] = e;
  gdn[i] = nn;
}

// generic fixed-nnz ELL spmm: out[b,c,r] = sum_k in[b,c,cols[r,k]] * vals[r,k]
__global__ void spmm_ell_kernel(const float* __restrict__ x,
                                const int* __restrict__ cols,
                                const float* __restrict__ vals,
                                float* __restrict__ out,
                                int R, int K, int Vin) {
  int i = blockIdx.x * blockDim.x + threadIdx.x;
  const int total = BB * CH * R;
  if (i >= total) return;
  int r  = i % R;
  int bc = i / R;
  const float* xb = x + (long)bc * Vin;
  float acc = 0.f;
  for (int k = 0; k < K; ++k)
    acc += xb[cols[r * K + k]] * vals[r * K + k];
  out[i] = acc;
}

// -------------------------------------------------- mesh-conv dense contraction
// out[b,o,v] = sum_{i,k} feat_k[b,i,v] * coeffs[o,i,k] + bias[o]
// WMMA f32 16x16x4: per step one input channel i, K-slice = 4 mesh features.
__global__ __launch_bounds__(256) void meshconv_gemm_kernel(
    const float* __restrict__ fx,  const float* __restrict__ flap,
    const float* __restrict__ fge, const float* __restrict__ fgn,
    const float* __restrict__ coeffs, const float* __restrict__ bias,
    float* __restrict__ out) {
  __shared__ float sB[OC * CH * 4];  // 16384 floats = 64KB, fits 320KB WGP LDS
  for (int t = threadIdx.x; t < OC * CH * 4; t += 256) sB[t] = coeffs[t];
  __syncthreads();

  const int wave  = threadIdx.x >> 5;
  const int lane  = threadIdx.x & 31;
  const int mtile = blockIdx.x * WPB + wave;
  if (mtile >= MTILES) return;  // wave-uniform exit: EXEC stays all-ones below

  const int bidx  = blockIdx.y;
  const int m     = lane & 15;
  const int hpart = lane >> 4;       // lane half selects K-pair {0,1} vs {2,3}
  const int v0    = mtile * 16;
  int v = v0 + m; if (v >= NV) v = NV - 1;   // clamp (masked store below)
  const float* p0 = hpart ? fge : fx;        // k = 2*hpart
  const float* p1 = hpart ? fgn : flap;      // k = 2*hpart + 1
  const long  base = (long)bidx * CH * NV + v;
  const int   n    = lane & 15;
  const int   k0   = 2 * hpart;

  v8f acc0 = {}, acc1 = {}, acc2 = {}, acc3 = {};
  for (int i = 0; i < CH; ++i) {
    v2f a;
    a.x = p0[base + (long)i * NV];
    a.y = p1[base + (long)i * NV];
    v2f b0 = *(const v2f*)&sB[(0 * 16 + n) * 256 + i * 4 + k0];
    v2f b1 = *(const v2f*)&sB[(1 * 16 + n) * 256 + i * 4 + k0];
    v2f b2 = *(const v2f*)&sB[(2 * 16 + n) * 256 + i * 4 + k0];
    v2f b3 = *(const v2f*)&sB[(3 * 16 + n) * 256 + i * 4 + k0];
    acc0 = __builtin_amdgcn_wmma_f32_16x16x4_f32(false, a, false, b0, (short)0, acc0, false, false);
    acc1 = __builtin_amdgcn_wmma_f32_16x16x4_f32(false, a, false, b1, (short)0, acc1, false, false);
    acc2 = __builtin_amdgcn_wmma_f32_16x16x4_f32(false, a, false, b2, (short)0, acc2, false, false);
    acc3 = __builtin_amdgcn_wmma_f32_16x16x4_f32(false, a, false, b3, (short)0, acc3, false, false);
  }

  v8f accs[4] = {acc0, acc1, acc2, acc3};
#pragma unroll
  for (int t = 0; t < 4; ++t) {
    const int o  = t * 16 + n;
    const float bv = bias[o];
#pragma unroll
    for (int r = 0; r < 8; ++r) {
      const int row  = r + 8 * hpart;   // D VGPR r holds rows r and r+8
      const int vout = v0 + row;
      if (vout < NV)
        out[(long)bidx * OC * NV + (long)o * NV + vout] = accs[t][r] + bv;
    }
  }
}

// -------------------------------------------------- 1x1 conv (optionally concat)
__global__ __launch_bounds__(256) void pconv_gemm_kernel(
    const float* __restrict__ s1, const float* __restrict__ s2,
    int C1, int C2,
    const float* __restrict__ w, const float* __restrict__ bias,
    float* __restrict__ out) {
  __shared__ float sW[OC * 128];  // 32KB max
  const int K = C1 + C2;
  for (int t = threadIdx.x; t < OC * K; t += 256) sW[t] = w[t];
  __syncthreads();

  const int wave  = threadIdx.x >> 5;
  const int lane  = threadIdx.x & 31;
  const int mtile = blockIdx.x * WPB + wave;
  if (mtile >= MTILES) return;

  const int bidx  = blockIdx.y;
  const int m     = lane & 15;
  const int hpart = lane >> 4;
  const int v0    = mtile * 16;
  int v = v0 + m; if (v >= NV) v = NV - 1;
  const long base1 = (long)bidx * C1 * NV + v;
  const long base2 = (long)bidx * C2 * NV + v;
  const int  n     = lane & 15;
  const int  koff  = 2 * hpart;

  v8f acc0 = {}, acc1 = {}, acc2 = {}, acc3 = {};
  for (int c0 = 0; c0 < K; c0 += 4) {
    const int ca = c0 + koff;  // ca and ca+1 are on the same side of the concat
    v2f a;
    if (ca < C1) {
      const long bb = base1 + (long)ca * NV;
      a.x = s1[bb]; a.y = s1[bb + NV];
    } else {
      const long bb = base2 + (long)(ca - C1) * NV;
      a.x = s2[bb]; a.y = s2[bb + NV];
    }
    v2f b0 = *(const v2f*)&sW[(0 * 16 + n) * K + c0 + koff];
    v2f b1 = *(const v2f*)&sW[(1 * 16 + n) * K + c0 + koff];
    v2f b2 = *(const v2f*)&sW[(2 * 16 + n) * K + c0 + koff];
    v2f b3 = *(const v2f*)&sW[(3 * 16 + n) * K + c0 + koff];
    acc0 = __builtin_amdgcn_wmma_f32_16x16x4_f32(false, a, false, b0, (short)0, acc0, false, false);
    acc1 = __builtin_amdgcn_wmma_f32_16x16x4_f32(false, a, false, b1, (short)0, acc1, false, false);
    acc2 = __builtin_amdgcn_wmma_f32_16x16x4_f32(false, a, false, b2, (short)0, acc2, false, false);
    acc3 = __builtin_amdgcn_wmma_f32_16x16x4_f32(false, a, false, b3, (short)0, acc3, false, false);
  }

  v8f accs[4] = {acc0, acc1, acc2, acc3};
#pragma unroll
  for (int t = 0; t < 4; ++t) {
    const int o  = t * 16 + n;
    const float bv = bias[o];
#pragma unroll
    for (int r = 0; r < 8; ++r) {
      const int row  = r + 8 * hpart;
      const int vout = v0 + row;
      if (vout < NV)
        out[(long)bidx * OC * NV + (long)o * NV + vout] = accs[t][r] + bv;
    }
  }
}

// -------------------------------------------------- BatchNorm (training mode)
__global__ __launch_bounds__(256) void bn_stats_kernel(const float* __restrict__ x,
                                                       float* __restrict__ stats) {
  const int o = blockIdx.x;
  __shared__ float ssum[256], ssq[256];
  float sum = 0.f, sq = 0.f;
  for (int b = 0; b < BB; ++b) {
    const float* xb = x + (long)b * OC * NV + (long)o * NV;
    for (int v = threadIdx.x; v < NV; v += 256) {
      float val = xb[v];
      sum += val; sq += val * val;
    }
  }
  ssum[threadIdx.x] = sum; ssq[threadIdx.x] = sq;
  __syncthreads();
  for (int s = 128; s > 0; s >>= 1) {
    if (threadIdx.x < s) {
      ssum[threadIdx.x] += ssum[threadIdx.x + s];
      ssq[threadIdx.x]  += ssq[threadIdx.x + s];
    }
    __syncthreads();
  }
  if (threadIdx.x == 0) { stats[2 * o] = ssum[0]; stats[2 * o + 1] = ssq[0]; }
}

__global__ void bn_relu_kernel(float* __restrict__ x, const float* __restrict__ stats,
                               const float* __restrict__ g, const float* __restrict__ beta) {
  int i = blockIdx.x * blockDim.x + threadIdx.x;
  const int total = BB * OC * NV;
  if (i >= total) return;
  const int o = (i / NV) % OC;
  const float cnt  = (float)(BB * NV);
  const float mean = stats[2 * o] / cnt;
  const float var  = stats[2 * o + 1] / cnt - mean * mean;
  const float sc   = g[o] * rsqrtf(var + EPSV);
  const float y    = (x[i] - mean) * sc + beta[o];
  x[i] = y > 0.f ? y : 0.f;
}

__global__ void final_kernel(const float* __restrict__ h3, const float* __restrict__ s,
                             const float* __restrict__ st3, const float* __restrict__ sts,
                             const float* __restrict__ g3, const float* __restrict__ b3,
                             const float* __restrict__ gs, const float* __restrict__ bs,
                             float* __restrict__ out) {
  int i = blockIdx.x * blockDim.x + threadIdx.x;
  const int total = BB * OC * NV;
  if (i >= total) return;
  const int o = (i / NV) % OC;
  const float cnt = (float)(BB * NV);
  const float m3  = st3[2 * o] / cnt;
  const float v3  = st3[2 * o + 1] / cnt - m3 * m3;
  const float ms  = sts[2 * o] / cnt;
  const float vs  = sts[2 * o + 1] / cnt - ms * ms;
  const float a = g3[o] * (h3[i] - m3) * rsqrtf(v3 + EPSV) + b3[o];
  const float c = gs[o] * (s[i]  - ms) * rsqrtf(vs + EPSV) + bs[o];
  const float y = a + c;
  out[i] = y > 0.f ? y : 0.f;
}

// ---------------------------------------------------------------- launcher
extern "C" void kernel_launch(void* const* d_in, const int* in_sizes, int n_in,
                              void* d_out, int out_size, void* d_ws, size_t ws_size,
                              hipStream_t stream) {
  (void)in_sizes; (void)n_in; (void)out_size; (void)ws_size;
  const float* x1  = (const float*)d_in[0];
  const float* x2  = (const float*)d_in[1];
  const int*   Gc  = (const int*)  d_in[2];
  const float* Gv  = (const float*)d_in[3];
  const int*   Lc  = (const int*)  d_in[4];
  const float* Lv  = (const float*)d_in[5];
  const int*   Fc  = (const int*)  d_in[6];
  const float* Fv  = (const float*)d_in[7];
  const float* EW  = (const float*)d_in[8];
  const float* NS  = (const float*)d_in[9];
  const int*   v2p = (const int*)  d_in[10];
  const float* upc = (const float*)d_in[11];
  const float* upb = (const float*)d_in[12];
  const float* c2c = (const float*)d_in[13];
  const float* c2b = (const float*)d_in[14];
  const float* w1  = (const float*)d_in[15];
  const float* b1  = (const float*)d_in[16];
  const float* w3  = (const float*)d_in[17];
  const float* b3  = (const float*)d_in[18];
  const float* wS  = (const float*)d_in[19];
  const float* bS  = (const float*)d_in[20];
  const float* bn1g = (const float*)d_in[21];
  const float* bn1b = (const float*)d_in[22];
  const float* bn2g = (const float*)d_in[23];
  const float* bn2b = (const float*)d_in[24];
  const float* bn3g = (const float*)d_in[25];
  const float* bn3b = (const float*)d_in[26];
  const float* bnsg = (const float*)d_in[27];
  const float* bnsb = (const float*)d_in[28];

  const size_t SV = (size_t)BB * CH * NV;   // 5,243,136
  const size_t SF = (size_t)BB * CH * NF;   // 10,485,760
  float* W    = (float*)d_ws;
  float* xin  = W;            // also reused as h after first mesh conv
  float* gde  = xin + SV;
  float* gdn  = gde + SF;
  float* lap  = gdn + SF;
  float* ge   = lap + SV;
  float* gn   = ge  + SV;
  float* xu   = gn  + SV;
  float* sbuf = xu  + SV;
  float* h2   = sbuf + SV;
  float* h3b  = h2  + SV;
  float* st   = h3b + SV;     // 4 stat blocks of 2*OC floats

  const dim3 blk(256);
  const dim3 gV((unsigned)((SV + 255) / 256));
  const dim3 gF((unsigned)((SF + 255) / 256));
  const dim3 gP((unsigned)(((size_t)BB * CH * NVP + 255) / 256));
  const dim3 gG((MTILES + WPB - 1) / WPB, BB);

  // --- MeshConv_transpose: pad-with-ones, scatter x1, mesh conv -> xu
  fill_ones_kernel<<<gV, blk, 0, stream>>>(xin, (int)SV);
  scatter_x1_kernel<<<gP, blk, 0, stream>>>(x1, v2p, xin);
  face_grad_kernel<<<gF, blk, 0, stream>>>(xin, Gc, Gv, EW, NS, gde, gdn);
  spmm_ell_kernel<<<gV, blk, 0, stream>>>(xin, Lc, Lv, lap, NV, 7, NV);
  spmm_ell_kernel<<<gV, blk, 0, stream>>>(gde, Fc, Fv, ge, NV, 6, NF);
  spmm_ell_kernel<<<gV, blk, 0, stream>>>(gdn, Fc, Fv, gn, NV, 6, NF);
  meshconv_gemm_kernel<<<gG, blk, 0, stream>>>(xin, lap, ge, gn, upc, upb, xu);

  // --- x = concat([x2, xu]); conv1 and shortcut conv (xin buffer becomes h)
  pconv_gemm_kernel<<<gG, blk, 0, stream>>>(x2, xu, CH, CH, w1, b1, xin);
  pconv_gemm_kernel<<<gG, blk, 0, stream>>>(x2, xu, CH, CH, wS, bS, sbuf);
  bn_stats_kernel<<<dim3(OC), blk, 0, stream>>>(xin, st + 0);
  bn_relu_kernel<<<gV, blk, 0, stream>>>(xin, st + 0, bn1g, bn1b);

  // --- second mesh conv on h -> h2
  face_grad_kernel<<<gF, blk, 0, stream>>>(xin, Gc, Gv, EW, NS, gde, gdn);
  spmm_ell_kernel<<<gV, blk, 0, stream>>>(xin, Lc, Lv, lap, NV, 7, NV);
  spmm_ell_kernel<<<gV, blk, 0, stream>>>(gde, Fc, Fv, ge, NV, 6, NF);
  spmm_ell_kernel<<<gV, blk, 0, stream>>>(gdn, Fc, Fv, gn, NV, 6, NF);
  meshconv_gemm_kernel<<<gG, blk, 0, stream>>>(xin, lap, ge, gn, c2c, c2b, h2);
  bn_stats_kernel<<<dim3(OC), blk, 0, stream>>>(h2, st + 128);
  bn_relu_kernel<<<gV, blk, 0, stream>>>(h2, st + 128, bn2g, bn2b);

  // --- conv3, BN stats for both branches, fused bn+bn+add+relu -> d_out
  pconv_gemm_kernel<<<gG, blk, 0, stream>>>(h2, nullptr, OC, 0, w3, b3, h3b);
  bn_stats_kernel<<<dim3(OC), blk, 0, stream>>>(h3b, st + 256);
  bn_stats_kernel<<<dim3(OC), blk, 0, stream>>>(sbuf, st + 384);
  final_kernel<<<gV, blk, 0, stream>>>(h3b, sbuf, st + 256, st + 384,
                                       bn3g, bn3b, bnsg, bnsb, (float*)d_out);
}